// DeepLK_22531398435285
// MI455X (gfx1250) — compile-verified
//
#include <hip/hip_runtime.h>
#include <math.h>

#define H_IMG 512
#define W_IMG 512
#define BATCH 8
#define NPIX  (H_IMG * W_IMG)
#define N_ITR 10
#define TOL_F 1e-3f
#define CENTER 255.5f   /* (W-1)/2 = (H-1)/2 */

typedef __attribute__((ext_vector_type(2))) float v2f;
typedef __attribute__((ext_vector_type(8))) float v8f;

// Workspace layout (float offsets); total 1216 floats = 4864 bytes.
#define WS_HM   0      // B x 8 x 8 Gram accumulators
#define WS_INVH 512    // B x 8 x 8 inverse
#define WS_B    1024   // B x 8   b = J^T r accumulators
#define WS_P    1088   // B x 8   p
#define WS_DP   1152   // B x 8   previous dp (for gate)

// ---------------------------------------------------------------------------
// Zero/initialize workspace (d_ws is poisoned; must init every launch).
// ---------------------------------------------------------------------------
__global__ void init_ws(float* __restrict__ ws) {
  int t = threadIdx.x;
  if (t < 512) ws[WS_HM + t] = 0.0f;
  if (t < 64) {
    ws[WS_B  + t] = 0.0f;
    ws[WS_P  + t] = 0.0f;
    ws[WS_DP + t] = 1.0f;   // reference: dp0 = ones -> first gate passes
  }
}

// ---------------------------------------------------------------------------
// Gram matrix Hm[b] = sum_k J_k J_k^T  via V_WMMA_F32_16X16X4_F32 (syrk).
// Each wave processes 512 pixels, 4 per WMMA (K=4), 4 WMMAs per loop trip.
// Lane mapping per ISA 7.12.2:
//   A 16x4 f32: lanes 0-15 hold (M=lane, K=0..1) in v[0..1]; lanes 16-31 K=2..3.
//   B  4x16 f32: transposed mirror -> identical register image for a syrk.
// C/D 16x16 f32: VGPR m, lanes 0-7 => element (M=m, N=lane). Top-left 8x8 used.
// Jacobian column selection is branchless (per-lane one-hot masks hoisted),
// so the loop body runs with EXEC fully set - no exec-mask churn around WMMA.
// ---------------------------------------------------------------------------
__global__ __launch_bounds__(128) void gram_wmma(const float* __restrict__ temp,
                                                 float* __restrict__ ws) {
  const int b    = blockIdx.y;
  const int lane = threadIdx.x & 31;
  const int wave = threadIdx.x >> 5;
  const int col  = lane & 15;     // Jacobian column this lane carries (>=8 -> 0)
  const int half = lane >> 4;     // 0: K slots 0,1   1: K slots 2,3

  // One-hot column masks (hoisted; cols 8..15 get all zeros -> Jv == 0)
  const float m0 = (col == 0) ? 1.0f : 0.0f;
  const float m1 = (col == 1) ? 1.0f : 0.0f;
  const float m2 = (col == 2) ? 1.0f : 0.0f;
  const float m3 = (col == 3) ? 1.0f : 0.0f;
  const float m4 = (col == 4) ? 1.0f : 0.0f;
  const float m5 = (col == 5) ? 1.0f : 0.0f;
  const float m6 = (col == 6) ? 1.0f : 0.0f;
  const float m7 = (col == 7) ? 1.0f : 0.0f;

  const int PIX_PER_WAVE = 512;
  const int base = (blockIdx.x * 4 + wave) * PIX_PER_WAVE;
  const float* __restrict__ T = temp + (size_t)b * NPIX;

  v8f c = {};
  for (int k0 = base; k0 < base + PIX_PER_WAVE; k0 += 16) {
#pragma unroll
    for (int u = 0; u < 4; ++u) {          // 4 WMMAs per trip
      v2f a;
#pragma unroll
      for (int j = 0; j < 2; ++j) {        // the 2 K-slots this lane feeds
        const int k = k0 + u * 4 + half * 2 + j;
        const int y = k >> 9, x = k & (W_IMG - 1);
        const int xm = x > 0 ? x - 1 : 0, xp = x < W_IMG - 1 ? x + 1 : W_IMG - 1;
        const int ym = y > 0 ? y - 1 : 0, yp = y < H_IMG - 1 ? y + 1 : H_IMG - 1;
        const float gx = 0.5f * (T[y * W_IMG + xp] - T[y * W_IMG + xm]);
        const float gy = 0.5f * (T[yp * W_IMG + x] - T[ym * W_IMG + x]);
        const float X = (float)x - CENTER, Y = (float)y - CENTER;
        const float nXX = -X * X, nXY = -X * Y, nYY = -Y * Y;
        const float cA = m0 * X + m1 * Y + m2 + m6 * nXX + m7 * nXY;
        const float cB = m3 * X + m4 * Y + m5 + m6 * nXY + m7 * nYY;
        a[j] = cA * gx + cB * gy;
      }
      // D = A * B + C ; A == B register image (syrk), f32 WMMA, K=4.
      c = __builtin_amdgcn_wmma_f32_16x16x4_f32(false, a, false, a,
                                                (short)0, c, false, false);
    }
  }
  if (lane < 8) {
#pragma unroll
    for (int m = 0; m < 8; ++m)
      atomicAdd(&ws[WS_HM + b * 64 + m * 8 + lane], c[m]);
  }
}

// ---------------------------------------------------------------------------
// 8x8 Gauss-Jordan inverse per batch (Gram matrix is SPD; no pivoting).
// ---------------------------------------------------------------------------
__global__ void invert_kernel(float* __restrict__ ws) {
  const int b = threadIdx.x;
  if (b >= BATCH) return;
  float A[8][8], V[8][8];
  for (int i = 0; i < 8; ++i)
    for (int j = 0; j < 8; ++j) {
      A[i][j] = ws[WS_HM + b * 64 + i * 8 + j];
      V[i][j] = (i == j) ? 1.0f : 0.0f;
    }
  for (int k = 0; k < 8; ++k) {
    const float piv = 1.0f / A[k][k];
    for (int j = 0; j < 8; ++j) { A[k][j] *= piv; V[k][j] *= piv; }
    for (int r = 0; r < 8; ++r) {
      if (r == k) continue;
      const float f = A[r][k];
      for (int j = 0; j < 8; ++j) { A[r][j] -= f * A[k][j]; V[r][j] -= f * V[k][j]; }
    }
  }
  for (int i = 0; i < 8; ++i)
    for (int j = 0; j < 8; ++j)
      ws[WS_INVH + b * 64 + i * 8 + j] = V[i][j];
}

// ---------------------------------------------------------------------------
// One LK iteration: warp img with H(p), bilinear (zero OOB), residual vs
// masked temp, accumulate b = J^T r  (per-thread regs -> LDS atomics ->
// global f32 atomics).
// ---------------------------------------------------------------------------
__global__ __launch_bounds__(256) void residual_kernel(const float* __restrict__ img,
                                                       const float* __restrict__ temp,
                                                       float* __restrict__ ws) {
  const int b = blockIdx.y;
  __shared__ float sb[8];
  if (threadIdx.x < 8) sb[threadIdx.x] = 0.0f;
  __syncthreads();

  const float p0 = ws[WS_P + b * 8 + 0], p1 = ws[WS_P + b * 8 + 1];
  const float p2 = ws[WS_P + b * 8 + 2], p3 = ws[WS_P + b * 8 + 3];
  const float p4 = ws[WS_P + b * 8 + 4], p5 = ws[WS_P + b * 8 + 5];
  const float p6 = ws[WS_P + b * 8 + 6], p7 = ws[WS_P + b * 8 + 7];

  const float* __restrict__ I = img  + (size_t)b * NPIX;
  const float* __restrict__ T = temp + (size_t)b * NPIX;

  float acc0 = 0, acc1 = 0, acc2 = 0, acc3 = 0, acc4 = 0, acc5 = 0, acc6 = 0, acc7 = 0;
  const int stride = blockDim.x * gridDim.x;
  const float mx = 1.0f - 2.0f / (float)W_IMG;
  const float my = 1.0f - 2.0f / (float)H_IMG;

  for (int i = blockIdx.x * blockDim.x + threadIdx.x; i < NPIX; i += stride) {
    const int y = i >> 9, x = i & (W_IMG - 1);
    const float X = (float)x - CENTER, Y = (float)y - CENTER;
    // H(p) * [X;Y;1], projective divide, back to pixel coords
    const float xw = (1.0f + p0) * X + p1 * Y + p2;
    const float yw = p3 * X + (1.0f + p4) * Y + p5;
    const float wv = p6 * X + p7 * Y + 1.0f;
    const float iw = 1.0f / wv;
    const float Xw = xw * iw + CENTER;
    const float Yw = yw * iw + CENTER;

    // bilinear with zero padding
    const float x0f = floorf(Xw), y0f = floorf(Yw);
    const float wx1 = Xw - x0f,  wy1 = Yw - y0f;
    float samp = 0.0f;
#pragma unroll
    for (int t = 0; t < 4; ++t) {
      const float xi = x0f + (float)(t & 1);
      const float yi = y0f + (float)(t >> 1);
      const float wgt = ((t & 1) ? wx1 : 1.0f - wx1) * ((t >> 1) ? wy1 : 1.0f - wy1);
      const bool valid = (xi >= 0.0f) && (xi <= (float)(W_IMG - 1)) &&
                         (yi >= 0.0f) && (yi <= (float)(H_IMG - 1));
      const int xc = (int)fminf(fmaxf(xi, 0.0f), (float)(W_IMG - 1));
      const int yc = (int)fminf(fmaxf(yi, 0.0f), (float)(H_IMG - 1));
      const float v = I[yc * W_IMG + xc];
      samp += valid ? v * wgt : 0.0f;
    }

    const float xn = Xw / CENTER - 1.0f;
    const float yn = Yw / CENTER - 1.0f;
    const float mask = (xn > -mx && xn < mx && yn > -my && yn < my) ? 1.0f : 0.0f;
    const float r = samp - T[i] * mask;

    // template gradients (central diff, edge clamp) - L2-resident
    const int xm = x > 0 ? x - 1 : 0, xp = x < W_IMG - 1 ? x + 1 : W_IMG - 1;
    const int ym = y > 0 ? y - 1 : 0, yp = y < H_IMG - 1 ? y + 1 : H_IMG - 1;
    const float gx = 0.5f * (T[y * W_IMG + xp] - T[y * W_IMG + xm]);
    const float gy = 0.5f * (T[yp * W_IMG + x] - T[ym * W_IMG + x]);

    acc0 += X * gx * r;  acc1 += Y * gx * r;  acc2 += gx * r;
    acc3 += X * gy * r;  acc4 += Y * gy * r;  acc5 += gy * r;
    acc6 += (-X * X * gx - X * Y * gy) * r;
    acc7 += (-X * Y * gx - Y * Y * gy) * r;
  }

  atomicAdd(&sb[0], acc0); atomicAdd(&sb[1], acc1);
  atomicAdd(&sb[2], acc2); atomicAdd(&sb[3], acc3);
  atomicAdd(&sb[4], acc4); atomicAdd(&sb[5], acc5);
  atomicAdd(&sb[6], acc6); atomicAdd(&sb[7], acc7);
  __syncthreads();
  if (threadIdx.x < 8)
    atomicAdd(&ws[WS_B + b * 8 + threadIdx.x], sb[threadIdx.x]);
}

// ---------------------------------------------------------------------------
// dp = gate(||dp_prev||>TOL) * (invH @ b), dp[6:8]=0 ; p -= dp ; b = 0.
// Launched with exactly 64 threads (8 batches x 8 params).
// ---------------------------------------------------------------------------
__global__ void update_kernel(float* __restrict__ ws) {
  const int t = threadIdx.x;
  const int b = t >> 3, i = t & 7;
  float nsq = 0.0f;
  for (int j = 0; j < 8; ++j) {
    const float d = ws[WS_DP + b * 8 + j];
    nsq += d * d;
  }
  const float gate = (sqrtf(nsq) > TOL_F) ? 1.0f : 0.0f;
  float dpn = 0.0f;
  if (i < 6) {
    for (int j = 0; j < 8; ++j)
      dpn += ws[WS_INVH + b * 64 + i * 8 + j] * ws[WS_B + b * 8 + j];
  }
  dpn *= gate;
  __syncthreads();                 // all reads of dp/b done before writes
  ws[WS_P  + b * 8 + i] -= dpn;
  ws[WS_DP + b * 8 + i]  = dpn;
  ws[WS_B  + b * 8 + i]  = 0.0f;   // ready for next iteration
}

// ---------------------------------------------------------------------------
// Pack outputs: p [B,8,1] (64 floats) then H = I3 + [p,0].reshape(3,3) (72).
// ---------------------------------------------------------------------------
__global__ void finalize_kernel(const float* __restrict__ ws, float* __restrict__ out) {
  const int t = threadIdx.x;
  if (t < 64) {
    out[t] = ws[WS_P + t];
  } else if (t < 64 + BATCH * 9) {
    const int q = t - 64;
    const int b = q / 9, e = q % 9;
    float v = (e < 8) ? ws[WS_P + b * 8 + e] : 0.0f;
    if (e == 0 || e == 4 || e == 8) v += 1.0f;
    out[64 + q] = v;
  }
}

// ---------------------------------------------------------------------------
extern "C" void kernel_launch(void* const* d_in, const int* in_sizes, int n_in,
                              void* d_out, int out_size, void* d_ws, size_t ws_size,
                              hipStream_t stream) {
  const float* img  = (const float*)d_in[0];   // [8,1,512,512] f32
  const float* temp = (const float*)d_in[1];   // [8,1,512,512] f32
  float* ws  = (float*)d_ws;
  float* out = (float*)d_out;                  // 64 (p) + 72 (H) floats

  init_ws<<<1, 512, 0, stream>>>(ws);

  dim3 ggrid(128, BATCH);                      // 512 waves/batch * 512 px/wave
  gram_wmma<<<ggrid, 128, 0, stream>>>(temp, ws);

  invert_kernel<<<1, 32, 0, stream>>>(ws);

  dim3 rgrid(128, BATCH);
  for (int it = 0; it < N_ITR; ++it) {
    residual_kernel<<<rgrid, 256, 0, stream>>>(img, temp, ws);
    update_kernel<<<1, 64, 0, stream>>>(ws);
  }

  finalize_kernel<<<1, 160, 0, stream>>>(ws, out);
}